// Label_Distance_Loss_27118423507485
// MI455X (gfx1250) — compile-verified
//
#include <hip/hip_runtime.h>
#include <hip/hip_bf16.h>
#include <math.h>

// Problem constants (match reference)
#define NB_  8
#define NC_  4
#define NH_  96
#define NW_  96
#define HW_  (NH_ * NW_)
#define TILES_ ((HW_ + 15) / 16)   // 576 A-tiles per batch (upper bound)

typedef float v2f __attribute__((ext_vector_type(2)));
typedef float v8f __attribute__((ext_vector_type(8)));

// ---------------------------------------------------------------------------
// 0) zero the per-batch counters / accumulators
// ---------------------------------------------------------------------------
__global__ void ldl_init_kernel(int* aCnt, int* bCnt, float* sum) {
    int i = threadIdx.x;
    if (i < NB_) { aCnt[i] = 0; bCnt[i] = 0; sum[i] = 0.0f; }
}

// ---------------------------------------------------------------------------
// 1) pred = argmax_c x[b,c,h,w]  (first-max wins, like jnp.argmax)
// ---------------------------------------------------------------------------
__global__ void ldl_pred_kernel(const float* __restrict__ x, int* __restrict__ pred) {
    int idx = blockIdx.x * blockDim.x + threadIdx.x;
    if (idx >= NB_ * HW_) return;
    int b = idx / HW_;
    int p = idx - b * HW_;
    const float* xb = x + (size_t)b * NC_ * HW_ + p;
    float best = xb[0];
    int bi = 0;
#pragma unroll
    for (int c = 1; c < NC_; ++c) {
        float v = xb[c * HW_];
        if (v > best) { best = v; bi = c; }   // strict > keeps first max
    }
    pred[idx] = bi;
}

// ---------------------------------------------------------------------------
// 2) edge detect (3x3 box-sum != 9*center, borders excluded) and compact
//    A-list: pred edges with class != 0;  B-list: label edges with class == 1
// ---------------------------------------------------------------------------
__global__ void ldl_edge_kernel(const int* __restrict__ label,
                                const int* __restrict__ pred,
                                int* aCnt, int* bCnt,
                                float* aY, float* aX,
                                float* bY, float* bX) {
    int idx = blockIdx.x * blockDim.x + threadIdx.x;
    if (idx >= NB_ * HW_) return;
    int b = idx / HW_;
    int p = idx - b * HW_;
    int h = p / NW_;
    int w = p - h * NW_;
    if (h == 0 || h == NH_ - 1 || w == 0 || w == NW_ - 1) return;  // borders: never edges

    const int* pr = pred  + b * HW_;
    const int* lb = label + b * HW_;
    int pc = pr[p];
    int lc = lb[p];
    int ps = 0, ls = 0;
#pragma unroll
    for (int dy = -1; dy <= 1; ++dy)
#pragma unroll
        for (int dx = -1; dx <= 1; ++dx) {
            int q = p + dy * NW_ + dx;
            ps += pr[q];
            ls += lb[q];
        }
    // integer sums <= 27: exactly equivalent to reference's float "s == 9*m" test
    if (ps != 9 * pc && pc != 0) {
        int pos = atomicAdd(&aCnt[b], 1);
        aY[b * HW_ + pos] = (float)h;
        aX[b * HW_ + pos] = (float)w;
    }
    if (ls != 9 * lc && lc == 1) {
        int pos = atomicAdd(&bCnt[b], 1);
        bY[b * HW_ + pos] = (float)h;
        bX[b * HW_ + pos] = (float)w;
    }
}

// ---------------------------------------------------------------------------
// 3) Pairwise min-distance via V_WMMA_F32_16X16X4_F32.
//    D^2 = [y_a, x_a, y_a^2+x_a^2, 1] . [-2y_b, -2x_b, 1, y_b^2+x_b^2]
//    One wave32 per (batch, 16-row A-tile); loop B in 16-col chunks.
//    16x4 f32 A layout: lanes 0-15 = M, VGPR0/1 = K0/K1; lanes 16-31 = K2/K3.
//    16x16 f32 D layout: VGPR r -> row r (lanes 0-15) / row r+8 (lanes 16-31).
// ---------------------------------------------------------------------------
__global__ void __launch_bounds__(32)
ldl_dist_kernel(const int* __restrict__ aCnt, const int* __restrict__ bCnt,
                const float* __restrict__ aY, const float* __restrict__ aX,
                const float* __restrict__ bY, const float* __restrict__ bX,
                float* __restrict__ sum) {
    int blk  = blockIdx.x;
    int b    = blk / TILES_;
    int tile = blk - b * TILES_;
    int nA = aCnt[b];
    int nB = bCnt[b];
    if (tile * 16 >= nA || nB == 0) return;   // wave-uniform exit

    int lane = threadIdx.x;
    int n  = lane & 15;     // column / A-row index within tile
    int hi = lane >> 4;     // lane half selects K pair (A) and rows r vs r+8 (D)

    // Build A operand: row m = tile*16 + n  (padded rows duplicate last valid;
    // they are masked out of the final sum)
    int am  = tile * 16 + n;
    int amc = (am < nA) ? am : (nA - 1);
    float ay = aY[b * HW_ + amc];
    float ax = aX[b * HW_ + amc];
    float as = ay * ay + ax * ax;          // <= 18050: exact in f32
    v2f Aop;
    Aop.x = hi ? as   : ay;                // K2 : K0
    Aop.y = hi ? 1.0f : ax;                // K3 : K1

    float mind[8];
#pragma unroll
    for (int r = 0; r < 8; ++r) mind[r] = 3.0e38f;

    int chunks = (nB + 15) >> 4;
    for (int c = 0; c < chunks; ++c) {
        int bn  = c * 16 + n;
        int bnc = (bn < nB) ? bn : (nB - 1);     // clamped load (no OOB)
        float by = bY[b * HW_ + bnc];
        float bx = bX[b * HW_ + bnc];
        if (bn >= nB) { by = 1.0e4f; bx = 1.0e4f; }   // pad col -> D^2 ~ 2e8, never wins min
        float bs = by * by + bx * bx;
        v2f Bop;
        Bop.x = hi ? 1.0f : (-2.0f * by);    // K2 : K0
        Bop.y = hi ? bs   : (-2.0f * bx);    // K3 : K1

        v8f acc = {};
        // (neg_a, A, neg_b, B, c_mod, C, reuse_a, reuse_b)
        acc = __builtin_amdgcn_wmma_f32_16x16x4_f32(
            false, Aop, false, Bop, (short)0, acc, false, false);

#pragma unroll
        for (int r = 0; r < 8; ++r)
            mind[r] = fminf(mind[r], acc[r]);  // running min over B chunks
    }

    // cross-lane min over the 16 columns (stays within each 16-lane half)
#pragma unroll
    for (int m = 1; m <= 8; m <<= 1) {
#pragma unroll
        for (int r = 0; r < 8; ++r)
            mind[r] = fminf(mind[r], __shfl_xor(mind[r], m, 32));
    }

    // lane 0 of each half writes rows hi*8 .. hi*8+7
    if (n == 0) {
        float local = 0.0f;
#pragma unroll
        for (int r = 0; r < 8; ++r) {
            int row = hi * 8 + r;
            if (tile * 16 + row < nA)
                local += sqrtf(fmaxf(mind[r], 0.0f));
        }
        atomicAdd(&sum[b], local);
    }
}

// ---------------------------------------------------------------------------
// 4) finalize: mean over batches of (nA>0 && nB>0 ? sum/nA : 0)
// ---------------------------------------------------------------------------
__global__ void ldl_fin_kernel(const int* __restrict__ aCnt,
                               const int* __restrict__ bCnt,
                               const float* __restrict__ sum,
                               float* __restrict__ out) {
    if (threadIdx.x == 0 && blockIdx.x == 0) {
        float acc = 0.0f;
#pragma unroll
        for (int b = 0; b < NB_; ++b) {
            if (aCnt[b] > 0 && bCnt[b] > 0)
                acc += sum[b] / (float)aCnt[b];
        }
        out[0] = acc / (float)NB_;
    }
}

// ---------------------------------------------------------------------------
extern "C" void kernel_launch(void* const* d_in, const int* in_sizes, int n_in,
                              void* d_out, int out_size, void* d_ws, size_t ws_size,
                              hipStream_t stream) {
    const float* x     = (const float*)d_in[0];   // (B,C,H,W) f32
    const int*   label = (const int*)d_in[1];     // (B,H,W) i32
    float* out = (float*)d_out;                   // scalar f32

    // workspace layout
    char* ws = (char*)d_ws;
    int*   aCnt = (int*)ws;                       ws += NB_ * sizeof(int);
    int*   bCnt = (int*)ws;                       ws += NB_ * sizeof(int);
    float* sum  = (float*)ws;                     ws += NB_ * sizeof(float);
    int*   pred = (int*)ws;                       ws += (size_t)NB_ * HW_ * sizeof(int);
    float* aY   = (float*)ws;                     ws += (size_t)NB_ * HW_ * sizeof(float);
    float* aX   = (float*)ws;                     ws += (size_t)NB_ * HW_ * sizeof(float);
    float* bY   = (float*)ws;                     ws += (size_t)NB_ * HW_ * sizeof(float);
    float* bX   = (float*)ws;                     ws += (size_t)NB_ * HW_ * sizeof(float);

    const int npix = NB_ * HW_;
    const int blk256 = (npix + 255) / 256;

    ldl_init_kernel<<<1, 32, 0, stream>>>(aCnt, bCnt, sum);
    ldl_pred_kernel<<<blk256, 256, 0, stream>>>(x, pred);
    ldl_edge_kernel<<<blk256, 256, 0, stream>>>(label, pred, aCnt, bCnt, aY, aX, bY, bX);
    ldl_dist_kernel<<<NB_ * TILES_, 32, 0, stream>>>(aCnt, bCnt, aY, aX, bY, bX, sum);
    ldl_fin_kernel<<<1, 32, 0, stream>>>(aCnt, bCnt, sum, out);
}